// StockPredictor_11158325035658
// MI455X (gfx1250) — compile-verified
//
#include <hip/hip_runtime.h>

// ---------------------------------------------------------------------------
// Batched tiny-RNN scan on MI455X (gfx1250) via V_WMMA_F32_16X16X4_F32.
//
// Mapping (per wave, per chain of 16 batch elements):
//   A (16x4, constant)  = W_hh rows {0,1,2,3} placed at M rows {0,1,8,9}
//   B (4x16, recurrent) = h[k][b], k=K, b=N (batch along columns)
//   C (16x16)           = rows {0,1,8,9} hold x_t[b]*W_ih[j] + (b_ih+b_hh)[j]
//   D = A*B + C  ->  D.V0 = {h'[0] | h'[2]}, D.V1 = {h'[1] | h'[3]}
// which, after elementwise tanh, IS the next step's B operand (V0,V1):
// zero cross-lane shuffles per step; the matrix pipe carries the recurrence.
//
// Round-2 refinement: the C operand is a persistent accumulator fed back
// in-place (acc = wmma(A, h, acc)). Because the unused A rows are zero,
// D[r] == C[r] for the unused rows, so the six "zero" C registers are
// self-preserving — eliminating the per-step v_mov re-zeroing chains the
// register allocator emitted in round 1.
// ---------------------------------------------------------------------------

typedef __attribute__((ext_vector_type(2))) float v2f;
typedef __attribute__((ext_vector_type(8))) float v8f;

#define S_LEN 1024
#define HID 4

__device__ __forceinline__ float cdna5_tanh(float v) {
#if __has_builtin(__builtin_amdgcn_tanhf)
  return __builtin_amdgcn_tanhf(v);
#else
  float r;
  asm("v_tanh_f32 %0, %1" : "=v"(r) : "v"(v));
  return r;
#endif
}

__global__ void __launch_bounds__(128)
rnn_scan_wmma_kernel(const float* __restrict__ x,
                     const float* __restrict__ W_ih,
                     const float* __restrict__ W_hh,
                     const float* __restrict__ b_ih,
                     const float* __restrict__ b_hh,
                     const float* __restrict__ fc_w,
                     const float* __restrict__ fc_b,
                     float* __restrict__ out,
                     int Btot)
{
  const int gtid = blockIdx.x * blockDim.x + threadIdx.x;
  const int lane = threadIdx.x & 31;
  const int base = (gtid >> 5) << 5;     // 32 batch rows per wave (2 chains x 16)
  if (base >= Btot) return;              // wave-uniform: EXEC stays all-ones

  const int  m  = lane & 15;             // column (batch) index within tile
  const bool lo = lane < 16;

  // Per-lane constants: low half handles hidden dims {0,1}, high half {2,3}.
  const int jA = lo ? 0 : 2;
  const int jB = lo ? 1 : 3;
  const float wihA  = W_ih[jA];
  const float wihB  = W_ih[jB];
  const float biasA = b_ih[jA] + b_hh[jA];
  const float biasB = b_ih[jB] + b_hh[jB];
  const float fwA   = fc_w[jA];
  const float fwB   = fc_w[jB];
  const float fcb   = fc_b[0];

  // A operand: 16x4 f32 layout: lane l holds row M=l&15; V0 = K=(lo?0:2),
  // V1 = K=(lo?1:3). W_hh rows {0,1,2,3} go to M rows {0,1,8,9}, rest zero.
  v2f A = {0.f, 0.f};
  {
    const int row = (m == 0) ? 0 : (m == 1) ? 1 : (m == 8) ? 2 : (m == 9) ? 3 : -1;
    if (row >= 0) {
      const int k0 = lo ? 0 : 2;
      A.x = W_hh[row * HID + k0];
      A.y = W_hh[row * HID + k0 + 1];
    }
  }

  // Two independent chains per wave to hide WMMA->tanh latency.
  const float* __restrict__ xp0 = x + (size_t)(base + m)      * S_LEN;
  const float* __restrict__ xp1 = x + (size_t)(base + 16 + m) * S_LEN;

  v2f h0 = {0.f, 0.f};   // B operand, chain 0 (h starts at zero)
  v2f h1 = {0.f, 0.f};   // B operand, chain 1
  // Persistent in-place C/D accumulators. Rows 2..7 (and 10..15) are zero and
  // stay zero across wmma calls because the matching A rows are zero.
  v8f acc0 = {};
  v8f acc1 = {};

  // x register ring: 16 timesteps in flight per chain (covers L2 latency).
  float4 buf0[4], buf1[4];
#pragma unroll
  for (int i = 0; i < 4; ++i) {
    buf0[i] = *(const float4*)(xp0 + 4 * i);
    buf1[i] = *(const float4*)(xp1 + 4 * i);
  }

  for (int s = 0; s < S_LEN; s += 16) {
    const int sn = (s + 16) & (S_LEN - 1);   // wraps to 0 on last iter (harmless)
    float4 n0[4], n1[4];
#pragma unroll
    for (int i = 0; i < 4; ++i) {
      n0[i] = *(const float4*)(xp0 + sn + 4 * i);
      n1[i] = *(const float4*)(xp1 + sn + 4 * i);
    }

#pragma unroll
    for (int i = 0; i < 4; ++i) {
      const float xs0[4] = {buf0[i].x, buf0[i].y, buf0[i].z, buf0[i].w};
      const float xs1[4] = {buf1[i].x, buf1[i].y, buf1[i].z, buf1[i].w};
#pragma unroll
      for (int u = 0; u < 4; ++u) {
        // Input + bias injection: overwrite only lanes [0],[1] of the
        // persistent accumulator (off the wmma->tanh critical path).
        acc0[0] = __builtin_fmaf(xs0[u], wihA, biasA);
        acc0[1] = __builtin_fmaf(xs0[u], wihB, biasB);
        acc1[0] = __builtin_fmaf(xs1[u], wihA, biasA);
        acc1[1] = __builtin_fmaf(xs1[u], wihB, biasB);

        // D = W_hh * h + (x*W_ih + bias), in place: D==C register block.
        acc0 = __builtin_amdgcn_wmma_f32_16x16x4_f32(
            false, A, false, h0, (short)0, acc0, false, false);
        acc1 = __builtin_amdgcn_wmma_f32_16x16x4_f32(
            false, A, false, h1, (short)0, acc1, false, false);

        // tanh(D.V0/V1) is directly the next step's B operand.
        h0.x = cdna5_tanh(acc0[0]);
        h0.y = cdna5_tanh(acc0[1]);
        h1.x = cdna5_tanh(acc1[0]);
        h1.y = cdna5_tanh(acc1[1]);
      }
    }

#pragma unroll
    for (int i = 0; i < 4; ++i) { buf0[i] = n0[i]; buf1[i] = n1[i]; }
  }

  // FC epilogue: lanes<16 hold h[0],h[1]; lanes>=16 hold h[2],h[3] (same b=m).
  float p0 = __builtin_fmaf(h0.x, fwA, h0.y * fwB);
  float p1 = __builtin_fmaf(h1.x, fwA, h1.y * fwB);
  p0 += __shfl_xor(p0, 16, 32);
  p1 += __shfl_xor(p1, 16, 32);
  if (lo) {
    out[base + m]      = p0 + fcb;
    out[base + 16 + m] = p1 + fcb;
  }
}

extern "C" void kernel_launch(void* const* d_in, const int* in_sizes, int n_in,
                              void* d_out, int out_size, void* d_ws, size_t ws_size,
                              hipStream_t stream) {
  (void)n_in; (void)out_size; (void)d_ws; (void)ws_size;
  const float* x    = (const float*)d_in[0];
  const float* W_ih = (const float*)d_in[1];
  const float* W_hh = (const float*)d_in[2];
  const float* b_ih = (const float*)d_in[3];
  const float* b_hh = (const float*)d_in[4];
  const float* fc_w = (const float*)d_in[5];
  const float* fc_b = (const float*)d_in[6];
  float* out = (float*)d_out;

  const int B = in_sizes[0] / S_LEN;   // INPUT == 1
  const int block = 128;               // 4 waves, 128 batch elements per block
  const int grid  = (B + 127) / 128;
  rnn_scan_wmma_kernel<<<grid, block, 0, stream>>>(x, W_ih, W_hh, b_ih, b_hh,
                                                   fc_w, fc_b, out, B);
}